// Feature_82910048682169
// MI455X (gfx1250) — compile-verified
//
#include <hip/hip_runtime.h>

// ---------------------------------------------------------------------------
// DeepMD-style descriptor, fully fused for MI455X (gfx1250).
// One 256-thread workgroup per (batch, atom). Neighbors counting-sorted by
// type so each bucket shares one weight channel; dense layers run on
// v_wmma_f32_16x16x4_f32 (full f32 precision, matrix pipe). Weight tiles are
// zero-padded in LDS so WMMA fragment loads are unconditional ds_load_b64.
// tanh uses the CDNA5 hardware V_TANH_F32 transcendental when available.
// All intermediates live in LDS (~125 KB of the 320 KB/WGP pool).
// ---------------------------------------------------------------------------

#define NTYPES 6
#define NATOM  192
#define NBATCH 4
#define NNB    191          // neighbors per atom
#define H0N    25
#define H1N    50
#define H2N    100
#define K0P    28           // 25 padded to multiple of 4
#define K1P    52           // 50 padded to multiple of 4
#define W1PR   64           // 50 output rows padded to 4 N-tiles
#define W2PR   112          // 100 output rows padded to 7 N-tiles
#define BLOCK  256

typedef __attribute__((ext_vector_type(2))) float v2f;
typedef __attribute__((ext_vector_type(8))) float v8f;

// ---- LDS layout (float offsets) ----
#define OFF_DINV 0
#define OFF_ENVA (OFF_DINV + NNB)              // 191
#define OFF_T2   (OFF_ENVA + NNB * 3)          // 764   (16B aligned)
#define OFF_H2   (OFF_T2 + NNB * 4)            // 1528
#define OFF_W1P  (OFF_H2 + NNB * H2N)          // 20628
#define OFF_W2P  (OFF_W1P + W1PR * K0P)        // 22420
#define OFF_B0C  (OFF_W2P + W2PR * K1P)        // 28244
#define OFF_B1C  (OFF_B0C + NTYPES * H0N)      // 28394
#define OFF_B2C  (OFF_B1C + NTYPES * H1N)      // 28694
#define OFF_W0C  (OFF_B2C + NTYPES * H2N)      // 29294
#define OFF_H0T  (OFF_W0C + NTYPES * H0N)      // 29444
#define OFF_H1T  (OFF_H0T + 16 * K0P)          // 29892
#define OFF_RED  (OFF_H1T + 16 * K1P)          // 30724
#define OFF_T1   (OFF_RED + 8 * 12)            // 30820
#define SMEM_FLOATS (OFF_T1 + 12)              // 30832
#define SMEM_INTS   (NNB + 7 + NTYPES + NATOM) // 396
#define SMEM_BYTES  (SMEM_FLOATS * 4 + SMEM_INTS * 4)   // 124912 B

__device__ __forceinline__ float fast_tanh(float x) {
#if __has_builtin(__builtin_amdgcn_tanhf)
    return __builtin_amdgcn_tanhf(x);        // v_tanh_f32 (CDNA5 TRANS op)
#elif __has_builtin(__builtin_amdgcn_tanh_f32)
    return __builtin_amdgcn_tanh_f32(x);
#else
    return tanhf(x);
#endif
}

extern "C" __global__ void __launch_bounds__(BLOCK)
deepmd_desc_kernel(const float* __restrict__ coords,
                   const int*   __restrict__ atom_types,
                   const float* __restrict__ W0, const float* __restrict__ b0,
                   const float* __restrict__ W1, const float* __restrict__ b1,
                   const float* __restrict__ W2, const float* __restrict__ b2,
                   float* __restrict__ out)
{
    extern __shared__ float smem[];
    float* dinv  = smem + OFF_DINV;
    float* env_a = smem + OFF_ENVA;   // [m][3]
    float* t2s   = smem + OFF_T2;     // [m][4]
    float* h2    = smem + OFF_H2;     // [m][100]
    float* W1p   = smem + OFF_W1P;    // [64][28]  zero-padded current channel
    float* W2p   = smem + OFF_W2P;    // [112][52] zero-padded current channel
    float* b0c   = smem + OFF_B0C;    // [6][25]
    float* b1c   = smem + OFF_B1C;    // [6][50]
    float* b2c   = smem + OFF_B2C;    // [6][100]
    float* W0c   = smem + OFF_W0C;    // [6][25]
    float* h0t   = smem + OFF_H0T;    // [16][K0P] tile of layer0 output
    float* h1t   = smem + OFF_H1T;    // [16][K1P] tile of layer1 output
    float* red   = smem + OFF_RED;    // [8][12]
    float* t1s   = smem + OFF_T1;     // [3][4]
    int*   sidx  = (int*)(smem + SMEM_FLOATS);  // sorted neighbor index
    int*   boff  = sidx + NNB;                  // bucket offsets [7]
    int*   bcnt  = boff + 7;                    // bucket counts [6]
    int*   tlds  = bcnt + NTYPES;               // staged atom types [192]

    const int i    = blockIdx.x;   // atom
    const int b    = blockIdx.y;   // batch
    const int tid  = threadIdx.x;
    const int lane = tid & 31;
    const int wave = tid >> 5;

    const int ti    = atom_types[i];
    const int cbase = ti * NTYPES;

    // zero tile buffers + padded weight tiles once; pads stay zero forever
    for (int idx = tid; idx < 16 * K0P; idx += BLOCK) h0t[idx] = 0.f;
    for (int idx = tid; idx < 16 * K1P; idx += BLOCK) h1t[idx] = 0.f;
    for (int idx = tid; idx < W1PR * K0P; idx += BLOCK) W1p[idx] = 0.f;
    for (int idx = tid; idx < W2PR * K1P; idx += BLOCK) W2p[idx] = 0.f;
    if (tid < NATOM) tlds[tid] = atom_types[tid];

    // ---- geometry: dinv, env_a ----
    const float cx = coords[((size_t)b * NATOM + i) * 3 + 0];
    const float cy = coords[((size_t)b * NATOM + i) * 3 + 1];
    const float cz = coords[((size_t)b * NATOM + i) * 3 + 2];
    for (int m = tid; m < NNB; m += BLOCK) {
        int j = (m < i) ? m : m + 1;
        float dx = cx - coords[((size_t)b * NATOM + j) * 3 + 0];
        float dy = cy - coords[((size_t)b * NATOM + j) * 3 + 1];
        float dz = cz - coords[((size_t)b * NATOM + j) * 3 + 2];
        float r2  = dx * dx + dy * dy + dz * dz;
        float inv = 1.0f / sqrtf(r2);
        dinv[m] = inv;
        float inv2 = inv * inv;
        env_a[m * 3 + 0] = dx * inv2;
        env_a[m * 3 + 1] = dy * inv2;
        env_a[m * 3 + 2] = dz * inv2;
    }
    __syncthreads();   // tlds visible

    // ---- stable counting sort of neighbors by type (deterministic) ----
    if (tid < NTYPES) {
        int c = 0;
        for (int m = 0; m < NNB; ++m) {
            int j = (m < i) ? m : m + 1;
            if (tlds[j] == tid) c++;
        }
        bcnt[tid] = c;
    }
    __syncthreads();
    if (tid == 0) {
        int acc = 0;
        for (int t = 0; t < NTYPES; ++t) { boff[t] = acc; acc += bcnt[t]; }
        boff[NTYPES] = acc;
    }
    __syncthreads();
    if (tid < NTYPES) {
        int pos = boff[tid];
        for (int m = 0; m < NNB; ++m) {
            int j = (m < i) ? m : m + 1;
            if (tlds[j] == tid) sidx[pos++] = m;
        }
    }

    // ---- bias / W0 slabs for the 6 channels of this atom type ----
    for (int idx = tid; idx < NTYPES * H0N; idx += BLOCK) {
        b0c[idx] = b0[cbase * H0N + idx];
        W0c[idx] = W0[cbase * H0N + idx];   // W0 is (C,25,1)
    }
    for (int idx = tid; idx < NTYPES * H1N; idx += BLOCK) b1c[idx] = b1[cbase * H1N + idx];
    for (int idx = tid; idx < NTYPES * H2N; idx += BLOCK) b2c[idx] = b2[cbase * H2N + idx];
    __syncthreads();

    // WMMA f32 16x16x4 lane patterns (ISA 7.12.2):
    //  A 16x4:  lanes 0-15 hold M=lane, K={0,1}; lanes 16-31 K={2,3}
    //  B 4x16:  lanes 0-15 hold N=lane, K={0,1}; lanes 16-31 K={2,3}
    //  C/D:     N = lane&15, VGPR r -> M = r + 8*(lane>=16)
    const int mrow = lane & 15;
    const int kp   = (lane >> 4) << 1;   // 0 or 2
    const int mhi  = (lane >> 4) << 3;   // 0 or 8

    for (int t = 0; t < NTYPES; ++t) {
        const int c = cbase + t;
        // load current channel into zero-padded LDS tiles
        for (int idx = tid; idx < H1N * H0N; idx += BLOCK) {
            int o = idx / H0N, k = idx - o * H0N;
            W1p[o * K0P + k] = W1[(size_t)c * H1N * H0N + idx];
        }
        for (int idx = tid; idx < H2N * H1N; idx += BLOCK) {
            int o = idx / H1N, k = idx - o * H1N;
            W2p[o * K1P + k] = W2[(size_t)c * H2N * H1N + idx];
        }
        // prefetch next channel's weights into cache while we compute
        if (t + 1 < NTYPES) {
            const float* nw2 = W2 + (size_t)(c + 1) * H2N * H1N;
            for (int p = tid; p < (H2N * H1N * 4) / 128; p += BLOCK)
                __builtin_prefetch(nw2 + p * 32, 0, 1);
            const float* nw1 = W1 + (size_t)(c + 1) * H1N * H0N;
            for (int p = tid; p < (H1N * H0N * 4 + 127) / 128; p += BLOCK)
                __builtin_prefetch(nw1 + p * 32, 0, 1);
        }
        __syncthreads();

        const int cnt    = bcnt[t];
        const int base   = boff[t];
        const int ntiles = (cnt + 15) >> 4;
        for (int tile = 0; tile < ntiles; ++tile) {
            // ---- layer 0: h0 = tanh(dinv * W0 + b0), scalar VALU/TRANS ----
            for (int idx = tid; idx < 16 * H0N; idx += BLOCK) {
                int r = idx / H0N, o = idx - r * H0N;
                int mloc = tile * 16 + r;
                float x = (mloc < cnt) ? dinv[sidx[base + mloc]] : 0.f;
                h0t[r * K0P + o] = fast_tanh(x * W0c[t * H0N + o] + b0c[t * H0N + o]);
            }
            __syncthreads();

            // ---- layer 1: (16x25)x(25x50) via WMMA, 4 N-tiles on waves 0-3 ----
            if (wave < 4) {
                const int n0 = wave * 16;
                v8f acc = {};
#pragma unroll
                for (int ks = 0; ks < K0P / 4; ++ks) {
                    const int k0 = ks * 4;
                    v2f a  = *(const v2f*)(h0t + mrow * K0P + k0 + kp);
                    v2f bb = *(const v2f*)(W1p + (n0 + mrow) * K0P + k0 + kp);
                    acc = __builtin_amdgcn_wmma_f32_16x16x4_f32(
                              false, a, false, bb, (short)0, acc, false, false);
                }
                const int o = n0 + mrow;
                if (o < H1N) {
                    const float bias = b1c[t * H1N + o];
                    const int   orc  = (o < H0N) ? o : o - H0N;
#pragma unroll
                    for (int r = 0; r < 8; ++r) {
                        int m = r + mhi;
                        h1t[m * K1P + o] = fast_tanh(acc[r] + bias) + h0t[m * K0P + orc];
                    }
                }
            }
            __syncthreads();

            // ---- layer 2: (16x50)x(50x100) via WMMA, 7 N-tiles on waves 0-6 ----
            for (int nt = wave; nt < 7; nt += 8) {
                const int n0 = nt * 16;
                v8f acc = {};
#pragma unroll
                for (int ks = 0; ks < K1P / 4; ++ks) {
                    const int k0 = ks * 4;
                    v2f a  = *(const v2f*)(h1t + mrow * K1P + k0 + kp);
                    v2f bb = *(const v2f*)(W2p + (n0 + mrow) * K1P + k0 + kp);
                    acc = __builtin_amdgcn_wmma_f32_16x16x4_f32(
                              false, a, false, bb, (short)0, acc, false, false);
                }
                const int o = n0 + mrow;
                if (o < H2N) {
                    const float bias = b2c[t * H2N + o];
                    const int   orc  = (o < H1N) ? o : o - H1N;
#pragma unroll
                    for (int r = 0; r < 8; ++r) {
                        int m = r + mhi;
                        int mloc = tile * 16 + m;
                        if (mloc < cnt) {
                            h2[sidx[base + mloc] * H2N + o] =
                                fast_tanh(acc[r] + bias) + h1t[m * K1P + orc];
                        }
                    }
                }
            }
            __syncthreads();
        }
        __syncthreads();   // safe before next bucket overwrites W1p/W2p
    }

    // ---- einsums: t1[3][4], t2[m][4], res[100][4] ----
    float part[12];
#pragma unroll
    for (int q = 0; q < 12; ++q) part[q] = 0.f;
    for (int m = tid; m < NNB; m += BLOCK) {
        float e0 = env_a[m * 3 + 0], e1 = env_a[m * 3 + 1], e2 = env_a[m * 3 + 2];
#pragma unroll
        for (int d = 0; d < 4; ++d) {
            float hv = h2[m * H2N + d];
            part[0 * 4 + d] += e0 * hv;
            part[1 * 4 + d] += e1 * hv;
            part[2 * 4 + d] += e2 * hv;
        }
    }
#pragma unroll
    for (int q = 0; q < 12; ++q) {      // deterministic wave32 tree reduction
        float v = part[q];
        for (int off = 16; off > 0; off >>= 1) v += __shfl_down(v, off, 32);
        if (lane == 0) red[wave * 12 + q] = v;
    }
    __syncthreads();
    if (tid < 12) {
        float v = 0.f;
        for (int w = 0; w < 8; ++w) v += red[w * 12 + tid];
        t1s[tid] = v;
    }
    __syncthreads();
    for (int m = tid; m < NNB; m += BLOCK) {
        float e0 = env_a[m * 3 + 0], e1 = env_a[m * 3 + 1], e2 = env_a[m * 3 + 2];
#pragma unroll
        for (int d = 0; d < 4; ++d)
            t2s[m * 4 + d] = e0 * t1s[0 * 4 + d] + e1 * t1s[1 * 4 + d] + e2 * t1s[2 * 4 + d];
    }
    __syncthreads();
    // res[D][d] = sum_m h2[m][D] * t2[m][d]; 200 threads: D x {m-chunk 0,1},
    // t2 rows via ds_load_b128, chunk-1 partials combined through LDS
    // (reusing dead h0t buffer), output via global_store_b128.
    float* rpart = h0t;                    // 448 floats >= 400, free after MLP
    const int chunk = tid / H2N;           // 0 or 1 for tid < 200
    const int Dr    = tid - chunk * H2N;
    float a0 = 0.f, a1 = 0.f, a2 = 0.f, a3 = 0.f;
    if (chunk < 2) {
        const int m0 = chunk ? 96 : 0;
        const int m1 = chunk ? NNB : 96;
        for (int m = m0; m < m1; ++m) {
            float  hv = h2[m * H2N + Dr];
            float4 tv = *(const float4*)(t2s + m * 4);
            a0 += hv * tv.x; a1 += hv * tv.y; a2 += hv * tv.z; a3 += hv * tv.w;
        }
        if (chunk == 1) {
            rpart[Dr * 4 + 0] = a0; rpart[Dr * 4 + 1] = a1;
            rpart[Dr * 4 + 2] = a2; rpart[Dr * 4 + 3] = a3;
        }
    }
    __syncthreads();
    if (chunk == 0) {
        float* outp = out + ((size_t)b * NATOM + i) * (H2N * 4);
        float4 r4 = make_float4(a0 + rpart[Dr * 4 + 0],
                                a1 + rpart[Dr * 4 + 1],
                                a2 + rpart[Dr * 4 + 2],
                                a3 + rpart[Dr * 4 + 3]);
        *(float4*)(outp + Dr * 4) = r4;
    }
}

extern "C" void kernel_launch(void* const* d_in, const int* in_sizes, int n_in,
                              void* d_out, int out_size, void* d_ws, size_t ws_size,
                              hipStream_t stream)
{
    (void)in_sizes; (void)n_in; (void)out_size; (void)d_ws; (void)ws_size;
    const float* coords = (const float*)d_in[0];
    const int*   at     = (const int*)d_in[1];
    const float* W0     = (const float*)d_in[2];
    const float* b0     = (const float*)d_in[3];
    const float* W1     = (const float*)d_in[4];
    const float* b1     = (const float*)d_in[5];
    const float* W2     = (const float*)d_in[6];
    const float* b2     = (const float*)d_in[7];
    float* out = (float*)d_out;

    // >64KB dynamic LDS: raise the limit (CDNA5 WGP has 320 KB)
    hipFuncSetAttribute((const void*)deepmd_desc_kernel,
                        hipFuncAttributeMaxDynamicSharedMemorySize, SMEM_BYTES);

    dim3 grid(NATOM, NBATCH);
    deepmd_desc_kernel<<<grid, BLOCK, SMEM_BYTES, stream>>>(
        coords, at, W0, b0, W1, b1, W2, b2, out);
}